// FPModule_24120536334939
// MI455X (gfx1250) — compile-verified
//
#include <hip/hip_runtime.h>
#include <hip/hip_bf16.h>

// Problem constants (match reference)
#define BB   16
#define NC   1024
#define NF   4096
#define CH   256     // feature channels
#define H1D  512     // hidden dim (2*CH -> 512)
#define NOUT 256
#define MTOT (BB * NF)   // 65536 fine points total

typedef float v2f __attribute__((ext_vector_type(2)));
typedef float v8f __attribute__((ext_vector_type(8)));

// ---------------------------------------------------------------------------
// Kernel 1: 3-NN inverse-distance interpolation.
// One block (256 threads) handles 256 fine points of one batch.
// Coarse positions (1024 x 3 fp32 = 12KB) staged in LDS; each thread keeps a
// running top-3 min over all 1024 candidates, then the whole block does a
// coalesced channel-parallel gather for each of its 256 points.
// ---------------------------------------------------------------------------
__global__ __launch_bounds__(256) void knn_interp_kernel(
    const float* __restrict__ x,         // [B*NC, CH]
    const float* __restrict__ pos,       // [B*NC, 3]
    const float* __restrict__ pos_skip,  // [B*NF, 3]
    float* __restrict__ interp)          // [B*NF, CH] (workspace)
{
    __shared__ float cpos[NC * 3];     // 12 KB
    __shared__ int   sidx[256 * 3];
    __shared__ float swgt[256 * 3];

    const int tid   = threadIdx.x;
    const int tiles = NF / 256;                 // 16 tiles per batch
    const int batch = blockIdx.x / tiles;
    const int tile  = blockIdx.x % tiles;

    // Stage coarse positions for this batch
    for (int i = tid; i < NC * 3; i += 256)
        cpos[i] = pos[(size_t)batch * NC * 3 + i];
    __syncthreads();

    const int fRow = batch * NF + tile * 256 + tid;   // global fine row
    const float px = pos_skip[(size_t)fRow * 3 + 0];
    const float py = pos_skip[(size_t)fRow * 3 + 1];
    const float pz = pos_skip[(size_t)fRow * 3 + 2];

    float d0 = 1e30f, d1 = 1e30f, d2 = 1e30f;
    int   i0 = 0,     i1 = 0,     i2 = 0;
    for (int j = 0; j < NC; ++j) {
        float dx = px - cpos[j * 3 + 0];
        float dy = py - cpos[j * 3 + 1];
        float dz = pz - cpos[j * 3 + 2];
        float d  = dx * dx + dy * dy + dz * dz;
        if (d < d0)      { d2 = d1; i2 = i1; d1 = d0; i1 = i0; d0 = d; i0 = j; }
        else if (d < d1) { d2 = d1; i2 = i1; d1 = d;  i1 = j; }
        else if (d < d2) { d2 = d;  i2 = j; }
    }
    float w0 = 1.0f / (d0 + 1e-16f);
    float w1 = 1.0f / (d1 + 1e-16f);
    float w2 = 1.0f / (d2 + 1e-16f);
    float inv = 1.0f / (w0 + w1 + w2);
    sidx[tid * 3 + 0] = i0;  swgt[tid * 3 + 0] = w0 * inv;
    sidx[tid * 3 + 1] = i1;  swgt[tid * 3 + 1] = w1 * inv;
    sidx[tid * 3 + 2] = i2;  swgt[tid * 3 + 2] = w2 * inv;
    __syncthreads();

    // Channel-parallel gather: thread tid owns channel tid for every point.
    const size_t rowBase = (size_t)batch * NF + (size_t)tile * 256;
    const float* xb = x + (size_t)batch * NC * CH;
    for (int p = 0; p < 256; ++p) {
        int   j0 = sidx[p * 3 + 0], j1 = sidx[p * 3 + 1], j2 = sidx[p * 3 + 2];
        float a0 = swgt[p * 3 + 0], a1 = swgt[p * 3 + 1], a2 = swgt[p * 3 + 2];
        float v  = a0 * xb[(size_t)j0 * CH + tid]
                 + a1 * xb[(size_t)j1 * CH + tid]
                 + a2 * xb[(size_t)j2 * CH + tid];
        interp[(rowBase + p) * CH + tid] = v;
    }
}

// ---------------------------------------------------------------------------
// Kernel 2: fused 2-layer MLP using V_WMMA_F32_16X16X4_F32.
// Block = 256 threads (8 waves), owns MB=32 rows.
//   Stage A = [interp | x_skip] (32 x 512 fp32) in LDS (stride 516 to avoid
//   bank conflicts). GEMM1: each wave owns 64 of the 512 hidden columns
//   (2 m-tiles x 4 n-tiles of v8f accumulators). ReLU+bias -> write h1 back
//   into the same LDS. GEMM2: each wave owns 32 of 256 output columns.
// ---------------------------------------------------------------------------
#define MB 32
#define LSTR (H1D + 4)   // 516 floats: 516 % 64 == 4 -> conflict-free rows

__global__ __launch_bounds__(256) void fp_mlp_kernel(
    const float* __restrict__ interp,   // [MTOT, CH]
    const float* __restrict__ x_skip,   // [MTOT, CH]
    const float* __restrict__ W1,       // [512, 512]
    const float* __restrict__ b1,       // [512]
    const float* __restrict__ W2,       // [512, 256]
    const float* __restrict__ b2,       // [256]
    float* __restrict__ out)            // [MTOT, 256]
{
    __shared__ float lds[MB * LSTR];    // ~66 KB, reused for A then h1

    const int tid     = threadIdx.x;
    const int wave    = tid >> 5;
    const int lane    = tid & 31;
    const int lane_lo = lane & 15;
    const int hi      = lane >> 4;          // 0: K=k,k+1  1: K=k+2,k+3
    const int row0    = blockIdx.x * MB;

    // ---- Stage A = [interp | x_skip] ----
    for (int idx = tid; idx < MB * H1D; idx += 256) {
        int r = idx >> 9;          // / 512
        int c = idx & (H1D - 1);
        float v = (c < CH) ? interp[(size_t)(row0 + r) * CH + c]
                           : x_skip[(size_t)(row0 + r) * CH + (c - CH)];
        lds[r * LSTR + c] = v;
    }
    __syncthreads();

    // ---- GEMM1: h1[32 x 512] = A @ W1 ----
    v8f acc1[2][4];
    for (int m = 0; m < 2; ++m)
        for (int n = 0; n < 4; ++n)
            acc1[m][n] = (v8f){0.f, 0.f, 0.f, 0.f, 0.f, 0.f, 0.f, 0.f};

    const int nbase1 = wave * 64;           // 4 column tiles of 16
    for (int k = 0; k < H1D; k += 4) {
        // A fragments (from LDS): a.x = A[row][k+2*hi], a.y = A[row][k+2*hi+1]
        v2f a[2];
        a[0] = *(const v2f*)&lds[(0 * 16 + lane_lo) * LSTR + k + 2 * hi];
        a[1] = *(const v2f*)&lds[(1 * 16 + lane_lo) * LSTR + k + 2 * hi];
        // B fragments (from global W1), column-striped across lanes
        v2f b[4];
        const int kk = k + 2 * hi;
        #pragma unroll
        for (int n = 0; n < 4; ++n) {
            const int col = nbase1 + n * 16 + lane_lo;
            b[n].x = W1[(size_t)kk * H1D + col];
            b[n].y = W1[(size_t)(kk + 1) * H1D + col];
        }
        #pragma unroll
        for (int m = 0; m < 2; ++m)
            #pragma unroll
            for (int n = 0; n < 4; ++n)
                acc1[m][n] = __builtin_amdgcn_wmma_f32_16x16x4_f32(
                    false, a[m], false, b[n], (short)0, acc1[m][n], false, false);
    }
    __syncthreads();   // everyone done reading A

    // ---- bias + ReLU, write h1 into LDS ----
    for (int m = 0; m < 2; ++m) {
        #pragma unroll
        for (int n = 0; n < 4; ++n) {
            const int col  = nbase1 + n * 16 + lane_lo;
            const float bv = b1[col];
            #pragma unroll
            for (int r = 0; r < 8; ++r) {
                const int row = m * 16 + r + hi * 8;
                lds[row * LSTR + col] = fmaxf(acc1[m][n][r] + bv, 0.0f);
            }
        }
    }
    __syncthreads();

    // ---- GEMM2: out[32 x 256] = h1 @ W2 ----
    v8f acc2[2][2];
    for (int m = 0; m < 2; ++m)
        for (int n = 0; n < 2; ++n)
            acc2[m][n] = (v8f){0.f, 0.f, 0.f, 0.f, 0.f, 0.f, 0.f, 0.f};

    const int nbase2 = wave * 32;           // 2 column tiles of 16
    for (int k = 0; k < H1D; k += 4) {
        v2f a[2];
        a[0] = *(const v2f*)&lds[(0 * 16 + lane_lo) * LSTR + k + 2 * hi];
        a[1] = *(const v2f*)&lds[(1 * 16 + lane_lo) * LSTR + k + 2 * hi];
        v2f b[2];
        const int kk = k + 2 * hi;
        #pragma unroll
        for (int n = 0; n < 2; ++n) {
            const int col = nbase2 + n * 16 + lane_lo;
            b[n].x = W2[(size_t)kk * NOUT + col];
            b[n].y = W2[(size_t)(kk + 1) * NOUT + col];
        }
        #pragma unroll
        for (int m = 0; m < 2; ++m)
            #pragma unroll
            for (int n = 0; n < 2; ++n)
                acc2[m][n] = __builtin_amdgcn_wmma_f32_16x16x4_f32(
                    false, a[m], false, b[n], (short)0, acc2[m][n], false, false);
    }

    // ---- bias + ReLU, store to global ----
    for (int m = 0; m < 2; ++m) {
        #pragma unroll
        for (int n = 0; n < 2; ++n) {
            const int col  = nbase2 + n * 16 + lane_lo;
            const float bv = b2[col];
            #pragma unroll
            for (int r = 0; r < 8; ++r) {
                const int row = row0 + m * 16 + r + hi * 8;
                out[(size_t)row * NOUT + col] = fmaxf(acc2[m][n][r] + bv, 0.0f);
            }
        }
    }
}

// ---------------------------------------------------------------------------
// Launch: inputs are x, pos, batch, x_skip, pos_skip, batch_skip, W1,b1,W2,b2.
// Output tuple (h [65536x256] f32, pos_skip [65536x3] f32, batch_skip [65536]
// i32) is concatenated flat in d_out.
// ---------------------------------------------------------------------------
extern "C" void kernel_launch(void* const* d_in, const int* in_sizes, int n_in,
                              void* d_out, int out_size, void* d_ws, size_t ws_size,
                              hipStream_t stream) {
    const float* x          = (const float*)d_in[0];
    const float* pos        = (const float*)d_in[1];
    const float* x_skip     = (const float*)d_in[3];
    const float* pos_skip   = (const float*)d_in[4];
    const void*  batch_skip = d_in[5];
    const float* W1         = (const float*)d_in[6];
    const float* b1         = (const float*)d_in[7];
    const float* W2         = (const float*)d_in[8];
    const float* b2         = (const float*)d_in[9];

    float* interp = (float*)d_ws;                 // [MTOT, CH] = 64 MB
    float* out    = (float*)d_out;

    // Phase 1: KNN + inverse-distance interpolation
    knn_interp_kernel<<<MTOT / 256, 256, 0, stream>>>(x, pos, pos_skip, interp);

    // Phase 2: fused (concat -> Linear(512) -> ReLU -> Linear(256) -> ReLU)
    fp_mlp_kernel<<<MTOT / MB, 256, 0, stream>>>(interp, x_skip, W1, b1, W2, b2, out);

    // Tuple outputs 2 & 3: pass-through copies (graph-capture safe D2D async)
    float* pos_out = out + (size_t)MTOT * NOUT;                 // 16,777,216
    void*  bat_out = (void*)(pos_out + (size_t)MTOT * 3);       // +196,608
    hipMemcpyAsync(pos_out, pos_skip, (size_t)MTOT * 3 * sizeof(float),
                   hipMemcpyDeviceToDevice, stream);
    hipMemcpyAsync(bat_out, batch_skip, (size_t)MTOT * sizeof(int),
                   hipMemcpyDeviceToDevice, stream);
}